// google_neural_machine_translation_model_51651276701898
// MI455X (gfx1250) — compile-verified
//
#include <hip/hip_runtime.h>
#include <cstddef>

// ---------------------------------------------------------------------------
// GNMT forward for MI455X (gfx1250, wave32). All matmuls via
// V_WMMA_F32_16X16X4_F32 (fp32 matrix pipe, matches fp32 reference numerics).
// Activations use native gfx1250 TRANS ops (v_tanh_f32 / v_exp_f32 / v_rcp_f32)
// which co-execute with the WMMA pipe.
// B=64, T=64, E=256, H=256, A=128, VOUT+1=32001.
// ---------------------------------------------------------------------------

typedef float v2f __attribute__((ext_vector_type(2)));
typedef float v8f __attribute__((ext_vector_type(8)));

__device__ __forceinline__ v8f wmma4(v2f a, v2f b, v8f c) {
  // (neg_a, A, neg_b, B, c_mod, C, reuse_a, reuse_b)
  return __builtin_amdgcn_wmma_f32_16x16x4_f32(false, a, false, b, (short)0, c,
                                               false, false);
}

__device__ __forceinline__ v8f v8zero() {
  v8f z = {0.f, 0.f, 0.f, 0.f, 0.f, 0.f, 0.f, 0.f};
  return z;
}

#define LOG2E 1.4426950408889634f

// exp via native v_exp_f32 (2^x)
__device__ __forceinline__ float fast_exp(float x) {
  return __builtin_amdgcn_exp2f(x * LOG2E);
}
// sigmoid(x) = rcp(1 + 2^(-x*log2e)); saturates correctly at +-inf
__device__ __forceinline__ float sigf(float x) {
  return __builtin_amdgcn_rcpf(1.f + __builtin_amdgcn_exp2f(-LOG2E * x));
}
// tanh: native V_TANH_F32 on gfx1250; exp2-based fallback otherwise
__device__ __forceinline__ float fast_tanh(float x) {
#if __has_builtin(__builtin_amdgcn_tanhf)
  return __builtin_amdgcn_tanhf(x);
#else
  float e = __builtin_amdgcn_exp2f(-2.f * LOG2E * x);  // e^-2x
  return (1.f - e) * __builtin_amdgcn_rcpf(1.f + e);
#endif
}

#define NBLK_SCAN 8

// Generation-counter grid barrier: monotonically increasing counter; no reset
// needed between barriers. Counter zeroed once per kernel_launch (memsetAsync).
__device__ __forceinline__ void grid_barrier(unsigned* bar) {
  __threadfence();   // release my writes to device scope
  __syncthreads();
  if (threadIdx.x == 0) {
    unsigned t = atomicAdd(bar, 1u);
    unsigned target = (t / (unsigned)NBLK_SCAN + 1u) * (unsigned)NBLK_SCAN;
    while (__hip_atomic_load(bar, __ATOMIC_RELAXED, __HIP_MEMORY_SCOPE_AGENT) <
           target) {
      __builtin_amdgcn_s_sleep(1);
    }
  }
  __syncthreads();
  __threadfence();   // acquire: see other blocks' writes
}

// ---------------------------------------------------------------------------
// Embedding gather: X[bt,:] = embed[tokens[bt],:]   (E=256)
// ---------------------------------------------------------------------------
__global__ __launch_bounds__(256) void embed_kernel(
    const int* __restrict__ tokens, const float* __restrict__ table,
    float* __restrict__ X) {
  int bt = blockIdx.x;
  X[(size_t)bt * 256 + threadIdx.x] =
      table[(size_t)tokens[bt] * 256 + threadIdx.x];
}

// ---------------------------------------------------------------------------
// Generic WMMA GEMM: C[M,N] = A[M,K] @ B + bias.  One 16x16 tile per wave.
// BT=0: B row-major [K,N] (ldb=N).  BT=1: B stored [N,K] row-major (ldb=K).
// Handles ragged N (projection N=32001) with guarded B loads / C stores.
// ---------------------------------------------------------------------------
template <int BT>
__global__ __launch_bounds__(256) void wmma_gemm_kernel(
    const float* __restrict__ A, const float* __restrict__ Bm,
    const float* __restrict__ bias, float* __restrict__ C, int M, int N, int K,
    int lda, int ldb, int ldc, int ntn) {
  const int lane = threadIdx.x & 31;
  const int wib = threadIdx.x >> 5;
  const int tile = blockIdx.x * 8 + wib;
  const int mt = tile / ntn;
  const int nt = tile - mt * ntn;
  if (mt * 16 >= M) return;

  const int l15 = lane & 15;
  const int hi = lane >> 4;
  const int k0b = hi << 1;
  const int arow = mt * 16 + l15;
  const int bcol = nt * 16 + l15;
  const bool bok = (bcol < N);

  v8f acc = v8zero();
#pragma unroll 8
  for (int kk = 0; kk < K; kk += 4) {
    const int k0 = kk + k0b;
    v2f a = *(const v2f*)&A[(size_t)arow * lda + k0];
    v2f b;
    if (BT) {
      b = *(const v2f*)&Bm[(size_t)bcol * ldb + k0];
    } else {
      b.x = bok ? Bm[(size_t)k0 * ldb + bcol] : 0.f;
      b.y = bok ? Bm[(size_t)(k0 + 1) * ldb + bcol] : 0.f;
    }
    acc = wmma4(a, b, acc);
  }
#pragma unroll
  for (int r = 0; r < 8; ++r) {
    int row = mt * 16 + r + hi * 8;
    if (bok) {
      float v = acc[r] + (bias ? bias[bcol] : 0.f);
      C[(size_t)row * ldc + bcol] = v;
    }
  }
}

// ---------------------------------------------------------------------------
// Persistent LSTM scan (one launch per layer). Grid = 8 blocks x 256 thr
// (64 waves). Wave gw owns h/c tile rows [mt*16,mt*16+16), cols [nt*16,+16).
// c tile lives in VGPRs for the whole scan; h ping-pongs through Hbuf[2][64][256].
// z = Zin[:,t,:] + h @ Wr ; gate order i,f,g,o.
// ---------------------------------------------------------------------------
__global__ __launch_bounds__(256) void lstm_scan_kernel(
    const float* __restrict__ Zin,  // [64,64,1024] (x@Wk + b, all t)
    const float* __restrict__ Wr,   // [256,1024]
    const float* __restrict__ res,  // optional residual base (same layout as out)
    float* __restrict__ out,        // [64,64,ld] (base may be offset for concat)
    int ld,
    float* __restrict__ Hbuf,       // [2,64,256]
    const float* __restrict__ h0, const float* __restrict__ c0,
    float* __restrict__ hT, float* __restrict__ cT, int dir, unsigned* bar) {
  __shared__ float ldsH[16 * 256];
  const int lane = threadIdx.x & 31;
  const int wib = threadIdx.x >> 5;
  const int gw = blockIdx.x * 8 + wib;  // 0..63
  const int mt = gw >> 4;               // 0..3  (batch tile)
  const int nt = gw & 15;               // 0..15 (h-col tile)
  const int l15 = lane & 15;
  const int hi = lane >> 4;
  const int acol0 = hi << 1;
  const int ccol = nt * 16 + l15;  // 0..255

  // init c (registers) and Hbuf parity 0
  v8f c;
#pragma unroll
  for (int r = 0; r < 8; ++r) {
    int brow = mt * 16 + r + hi * 8;
    c[r] = c0 ? c0[(size_t)brow * 256 + ccol] : 0.f;
    Hbuf[(size_t)brow * 256 + ccol] = h0 ? h0[(size_t)brow * 256 + ccol] : 0.f;
  }
  grid_barrier(bar);

  v8f hlast = v8zero();
  for (int tt = 0; tt < 64; ++tt) {
    const int t = (dir > 0) ? tt : (63 - tt);
    const float* Hrd = Hbuf + (size_t)(tt & 1) * (64 * 256);
    float* Hwr = Hbuf + (size_t)((tt & 1) ^ 1) * (64 * 256);

    // stage this block's batch-tile of h into LDS (rows mt*16..+16, all 256 k)
    {
      const float4* src = (const float4*)(Hrd + (size_t)mt * 16 * 256);
      float4* dst = (float4*)ldsH;
      for (int i = threadIdx.x; i < 1024; i += 256) dst[i] = src[i];
    }
    __syncthreads();

    v8f z[4];
    z[0] = v8zero(); z[1] = v8zero(); z[2] = v8zero(); z[3] = v8zero();
    const float* wb = Wr + ccol;  // + g*256 + k*1024
#pragma unroll 4
    for (int kk = 0; kk < 256; kk += 4) {
      const int k0 = kk + acol0;
      v2f a = *(const v2f*)&ldsH[l15 * 256 + k0];
      const float* wp = wb + (size_t)k0 * 1024;
#pragma unroll
      for (int g = 0; g < 4; ++g) {
        v2f b;
        b.x = wp[g * 256];
        b.y = wp[g * 256 + 1024];
        z[g] = wmma4(a, b, z[g]);
      }
    }

#pragma unroll
    for (int r = 0; r < 8; ++r) {
      int brow = mt * 16 + r + hi * 8;
      const float* zp = Zin + ((size_t)brow * 64 + t) * 1024 + ccol;
      float ig = sigf(z[0][r] + zp[0]);
      float fg = sigf(z[1][r] + zp[256]);
      float gg = fast_tanh(z[2][r] + zp[512]);
      float og = sigf(z[3][r] + zp[768]);
      float cn = fg * c[r] + ig * gg;
      c[r] = cn;
      float h = og * fast_tanh(cn);
      hlast[r] = h;
      Hwr[(size_t)brow * 256 + ccol] = h;
      size_t oidx = ((size_t)brow * 64 + t) * ld + ccol;
      out[oidx] = h + (res ? res[oidx] : 0.f);
    }
    grid_barrier(bar);
  }

  if (hT || cT) {
#pragma unroll
    for (int r = 0; r < 8; ++r) {
      int brow = mt * 16 + r + hi * 8;
      if (hT) hT[(size_t)brow * 256 + ccol] = hlast[r];
      if (cT) cT[(size_t)brow * 256 + ccol] = c[r];
    }
  }
}

// ---------------------------------------------------------------------------
// Decoder layer 1: per-step Bahdanau attention + LSTM cell. Persistent kernel,
// 8 blocks x 256 thr. Three phases per step, grid-synced:
//   A) q = h @ W_a^T (32 waves, WMMA)       -> Qbuf[64,128]
//   B) score/softmax/ctx (1 wave per batch) -> Ctxbuf[64,256], ctxout
//   C) z = Zx[:,t] + ctx@Wk2 + h@Wr (WMMA), gates -> h (ping-pong), d1out
// ---------------------------------------------------------------------------
__global__ __launch_bounds__(256) void decoder_scan_kernel(
    const float* __restrict__ Zx,    // [64,64,1024] = enc@Wk1 + b
    const float* __restrict__ Wk2,   // dec1_Wk rows 256..511, ldb=1024
    const float* __restrict__ Wr,    // [256,1024]
    const float* __restrict__ enc,   // [64,64,256]
    const float* __restrict__ Uenc,  // [64,64,128]
    const float* __restrict__ Wa,    // [128,256]
    const float* __restrict__ Va,    // [128]
    const float* __restrict__ h0, const float* __restrict__ c0,  // [64,256]
    float* __restrict__ d1out,   // base, ld 512 (concat slot 0)
    float* __restrict__ ctxout,  // base+256, ld 512 (concat slot 1)
    float* __restrict__ Hbuf,    // [2,64,256]
    float* __restrict__ Qbuf,    // [64,128]
    float* __restrict__ Ctxbuf,  // [64,256]
    unsigned* bar) {
  const int lane = threadIdx.x & 31;
  const int wib = threadIdx.x >> 5;
  const int gw = blockIdx.x * 8 + wib;  // 0..63
  const int mt = gw >> 4;
  const int nt = gw & 15;
  const int l15 = lane & 15;
  const int hi = lane >> 4;
  const int acol0 = hi << 1;
  const int ccol = nt * 16 + l15;

  v8f c;
#pragma unroll
  for (int r = 0; r < 8; ++r) {
    int brow = mt * 16 + r + hi * 8;
    c[r] = c0[(size_t)brow * 256 + ccol];
    Hbuf[(size_t)brow * 256 + ccol] = h0[(size_t)brow * 256 + ccol];
  }
  grid_barrier(bar);

  for (int t = 0; t < 64; ++t) {
    const float* Hrd = Hbuf + (size_t)(t & 1) * (64 * 256);
    float* Hwr = Hbuf + (size_t)((t & 1) ^ 1) * (64 * 256);

    // ---- Phase A: q = h @ W_a^T  (M=64, N=128, K=256) : 32 tiles
    if (gw < 32) {
      const int mt2 = gw >> 3;     // 0..3
      const int nt2 = gw & 7;      // 0..7
      const int arow = mt2 * 16 + l15;
      const int bcol = nt2 * 16 + l15;  // attention dim 0..127
      v8f q = v8zero();
#pragma unroll 4
      for (int kk = 0; kk < 256; kk += 4) {
        const int k0 = kk + acol0;
        v2f a = *(const v2f*)&Hrd[(size_t)arow * 256 + k0];
        v2f b = *(const v2f*)&Wa[(size_t)bcol * 256 + k0];  // W_a[a,k] (NT)
        q = wmma4(a, b, q);
      }
#pragma unroll
      for (int r = 0; r < 8; ++r)
        Qbuf[(size_t)(mt2 * 16 + r + hi * 8) * 128 + bcol] = q[r];
    }
    grid_barrier(bar);

    // ---- Phase B: attention (wave gw handles batch b = gw)
    {
      const int b = gw;
      const float* qp = Qbuf + (size_t)b * 128;
      const float* u0 = Uenc + ((size_t)b * 64 + lane) * 128;
      const float* u1 = Uenc + ((size_t)b * 64 + lane + 32) * 128;
      float sc0 = 0.f, sc1 = 0.f;
#pragma unroll 8
      for (int a = 0; a < 128; ++a) {
        float qa = qp[a];
        float va = Va[a];
        sc0 += va * fast_tanh(qa + u0[a]);
        sc1 += va * fast_tanh(qa + u1[a]);
      }
      float m = fmaxf(sc0, sc1);
      for (int off = 16; off; off >>= 1) m = fmaxf(m, __shfl_xor(m, off, 32));
      float e0 = fast_exp(sc0 - m), e1 = fast_exp(sc1 - m);
      float s = e0 + e1;
      for (int off = 16; off; off >>= 1) s += __shfl_xor(s, off, 32);
      float rs = 1.f / s;
      float a0 = e0 * rs, a1 = e1 * rs;

      float acc[8];
#pragma unroll
      for (int j = 0; j < 8; ++j) acc[j] = 0.f;
      for (int ss = 0; ss < 64; ++ss) {
        float al = (ss < 32) ? __shfl(a0, ss, 32) : __shfl(a1, ss - 32, 32);
        const float* er = enc + ((size_t)b * 64 + ss) * 256 + lane;
#pragma unroll
        for (int j = 0; j < 8; ++j) acc[j] += al * er[j * 32];
      }
#pragma unroll
      for (int j = 0; j < 8; ++j) {
        Ctxbuf[(size_t)b * 256 + lane + j * 32] = acc[j];
        ctxout[((size_t)b * 64 + t) * 512 + lane + j * 32] = acc[j];
      }
    }
    grid_barrier(bar);

    // ---- Phase C: z = Zx[:,t] + ctx@Wk2 + h@Wr ; gates; write h
    {
      v8f z[4];
      z[0] = v8zero(); z[1] = v8zero(); z[2] = v8zero(); z[3] = v8zero();
      const float* w2 = Wk2 + ccol;
      const float* wr = Wr + ccol;
#pragma unroll 2
      for (int kk = 0; kk < 256; kk += 4) {
        const int k0 = kk + acol0;
        v2f ac = *(const v2f*)&Ctxbuf[(size_t)(mt * 16 + l15) * 256 + k0];
        v2f ah = *(const v2f*)&Hrd[(size_t)(mt * 16 + l15) * 256 + k0];
        const float* p2 = w2 + (size_t)k0 * 1024;
        const float* pr = wr + (size_t)k0 * 1024;
#pragma unroll
        for (int g = 0; g < 4; ++g) {
          v2f b1, b2;
          b1.x = p2[g * 256];
          b1.y = p2[g * 256 + 1024];
          z[g] = wmma4(ac, b1, z[g]);
          b2.x = pr[g * 256];
          b2.y = pr[g * 256 + 1024];
          z[g] = wmma4(ah, b2, z[g]);
        }
      }
#pragma unroll
      for (int r = 0; r < 8; ++r) {
        int brow = mt * 16 + r + hi * 8;
        const float* zp = Zx + ((size_t)brow * 64 + t) * 1024 + ccol;
        float ig = sigf(z[0][r] + zp[0]);
        float fg = sigf(z[1][r] + zp[256]);
        float gg = fast_tanh(z[2][r] + zp[512]);
        float og = sigf(z[3][r] + zp[768]);
        float cn = fg * c[r] + ig * gg;
        c[r] = cn;
        float h = og * fast_tanh(cn);
        Hwr[(size_t)brow * 256 + ccol] = h;
        d1out[((size_t)brow * 64 + t) * 512 + ccol] = h;
      }
    }
    grid_barrier(bar);
  }
}

// ---------------------------------------------------------------------------
// Host orchestration
// ---------------------------------------------------------------------------
extern "C" void kernel_launch(void* const* d_in, const int* in_sizes, int n_in,
                              void* d_out, int out_size, void* d_ws,
                              size_t ws_size, hipStream_t stream) {
  const int* tokens = (const int*)d_in[0];
  const float* embed = (const float*)d_in[1];
  const float* enc1f_Wk = (const float*)d_in[2];
  const float* enc1f_Wr = (const float*)d_in[3];
  const float* enc1f_b = (const float*)d_in[4];
  const float* enc1b_Wk = (const float*)d_in[5];
  const float* enc1b_Wr = (const float*)d_in[6];
  const float* enc1b_b = (const float*)d_in[7];
  const float* enc2_Wk = (const float*)d_in[8];
  const float* enc2_Wr = (const float*)d_in[9];
  const float* enc2_b = (const float*)d_in[10];
  const float* enc_res_Wk = (const float*)d_in[11];
  const float* enc_res_Wr = (const float*)d_in[12];
  const float* enc_res_b = (const float*)d_in[13];
  const float* W_a = (const float*)d_in[14];
  const float* U_a = (const float*)d_in[15];
  const float* V_a = (const float*)d_in[16];
  const float* dec1_Wk = (const float*)d_in[17];
  const float* dec1_Wr = (const float*)d_in[18];
  const float* dec1_b = (const float*)d_in[19];
  const float* dec2_Wk = (const float*)d_in[20];
  const float* dec2_Wr = (const float*)d_in[21];
  const float* dec2_b = (const float*)d_in[22];
  const float* dec_res_Wk = (const float*)d_in[23];
  const float* dec_res_Wr = (const float*)d_in[24];
  const float* dec_res_b = (const float*)d_in[25];
  const float* out_W = (const float*)d_in[26];
  const float* out_b = (const float*)d_in[27];

  float* ws = (float*)d_ws;
  float* X = ws;                    // [4096,256]
  float* Zin = X + 1048576;         // [4096,1024]
  float* S0 = Zin + 4194304;        // [4096,512] concat buffer
  float* S1 = S0 + 2097152;         // [4096,256]
  float* S2 = S1 + 1048576;         // [4096,256]
  float* Uenc = S2 + 1048576;       // [4096,128]
  float* Hbuf = Uenc + 524288;      // [2,64,256]
  float* Hsave = Hbuf + 32768;      // [64,256] b_h
  float* Csave = Hsave + 16384;     // [64,256] b_c
  float* Qbuf = Csave + 16384;      // [64,128]
  float* Ctxb = Qbuf + 8192;        // [64,256]
  unsigned* bars = (unsigned*)(Ctxb + 16384);

  hipMemsetAsync(bars, 0, 256, stream);

  auto gemm_nn = [&](const float* A, const float* Bm, const float* bias,
                     float* C, int M, int N, int K) {
    int ntn = (N + 15) / 16;
    int tiles = (M / 16) * ntn;
    wmma_gemm_kernel<0><<<(tiles + 7) / 8, 256, 0, stream>>>(
        A, Bm, bias, C, M, N, K, K, N, N, ntn);
  };
  auto gemm_nt = [&](const float* A, const float* Bm, const float* bias,
                     float* C, int M, int N, int K) {
    int ntn = (N + 15) / 16;
    int tiles = (M / 16) * ntn;
    wmma_gemm_kernel<1><<<(tiles + 7) / 8, 256, 0, stream>>>(
        A, Bm, bias, C, M, N, K, K, K, N, ntn);
  };
  auto scan = [&](const float* zin, const float* wr, const float* res,
                  float* out, int ld, const float* h0, const float* c0,
                  float* hT, float* cT, int dir) {
    lstm_scan_kernel<<<NBLK_SCAN, 256, 0, stream>>>(zin, wr, res, out, ld, Hbuf,
                                                    h0, c0, hT, cT, dir, bars);
  };

  // Embedding
  embed_kernel<<<4096, 256, 0, stream>>>(tokens, embed, X);

  // Encoder layer 1 (bi-directional) -> S0 = [f_seq | b_seq_rev]  ([B,T,512])
  gemm_nn(X, enc1f_Wk, enc1f_b, Zin, 4096, 1024, 256);
  scan(Zin, enc1f_Wr, nullptr, S0, 512, nullptr, nullptr, nullptr, nullptr, +1);
  gemm_nn(X, enc1b_Wk, enc1b_b, Zin, 4096, 1024, 256);
  scan(Zin, enc1b_Wr, nullptr, S0 + 256, 512, nullptr, nullptr, Hsave, Csave, -1);

  // Encoder layer 2 (K=512) -> S1
  gemm_nn(S0, enc2_Wk, enc2_b, Zin, 4096, 1024, 512);
  scan(Zin, enc2_Wr, nullptr, S1, 256, nullptr, nullptr, nullptr, nullptr, +1);

  // Encoder residual layers 3..8
  float* cur = S1;
  float* alt = S2;
  for (int i = 0; i < 6; ++i) {
    gemm_nn(cur, enc_res_Wk + (size_t)i * 262144, enc_res_b + i * 1024, Zin,
            4096, 1024, 256);
    scan(Zin, enc_res_Wr + (size_t)i * 262144, cur, alt, 256, nullptr, nullptr,
         nullptr, nullptr, +1);
    float* tmp = cur; cur = alt; alt = tmp;
  }
  float* enc = cur;  // S1 after 6 swaps

  // Attention precompute: U_enc = enc @ U_a^T  (NT, N=128)
  gemm_nt(enc, U_a, nullptr, Uenc, 4096, 128, 256);
  // Zx = enc @ dec1_Wk[0:256,:] + dec1_b
  gemm_nn(enc, dec1_Wk, dec1_b, Zin, 4096, 1024, 256);

  // Decoder layer 1 with attention -> S0 = [d1 | ctxs]
  decoder_scan_kernel<<<NBLK_SCAN, 256, 0, stream>>>(
      Zin, dec1_Wk + 262144, dec1_Wr, enc, Uenc, W_a, V_a, Hsave, Csave, S0,
      S0 + 256, Hbuf, Qbuf, Ctxb, bars);

  // Decoder layer 2 (K=512) -> S1
  gemm_nn(S0, dec2_Wk, dec2_b, Zin, 4096, 1024, 512);
  scan(Zin, dec2_Wr, nullptr, S1, 256, nullptr, nullptr, nullptr, nullptr, +1);

  // Decoder residual layers 3..8
  cur = S1; alt = S2;
  for (int i = 0; i < 6; ++i) {
    gemm_nn(cur, dec_res_Wk + (size_t)i * 262144, dec_res_b + i * 1024, Zin,
            4096, 1024, 256);
    scan(Zin, dec_res_Wr + (size_t)i * 262144, cur, alt, 256, nullptr, nullptr,
         nullptr, nullptr, +1);
    float* tmp = cur; cur = alt; alt = tmp;
  }

  // Output projection: logits[4096,32001] = s @ out_W + out_b
  gemm_nn(cur, out_W, out_b, (float*)d_out, 4096, 32001, 256);
}